// MPNN_53352083751303
// MI455X (gfx1250) — compile-verified
//
#include <hip/hip_runtime.h>

// ---------------------------------------------------------------------------
// MPNN / NNConv on MI455X (gfx1250, wave32).
//
// Key algebraic factoring: W_e = ea[e]*W1 + B1  (edge MLP is Linear(1, D*D)),
// so   msg[e] = ea[e]*P1[src] + P2[src],  P1 = relu(h)@W1, P2 = relu(h)@B1.
// Per layer: one fused 12288x32 @ 32x96 fp32 WMMA GEMM (cols = W1|B1|root),
// then an edge fma+scatter, then finalize. Avoids the 512MB W_e tensor.
// ---------------------------------------------------------------------------

constexpr int D       = 32;
constexpr int NNODES  = 12288;    // B(64) * graph_size(192)
constexpr int NEDGES  = 131072;
constexpr int NLAYERS = 3;

typedef __attribute__((ext_vector_type(2))) float v2f;
typedef __attribute__((ext_vector_type(8))) float v8f;

// h0[n,d] = x[n]*ne_w[d] + ne_b[d]   (node_dim == 1, same embed for u and v)
__global__ void embed_kernel(const float* __restrict__ x,
                             const float* __restrict__ ne_w,
                             const float* __restrict__ ne_b,
                             float* __restrict__ h0) {
    int idx = blockIdx.x * blockDim.x + threadIdx.x;
    if (idx < NNODES * D) {
        int n = idx >> 5, d = idx & 31;
        h0[idx] = fmaf(x[n], ne_w[d], ne_b[d]);
    }
}

__global__ void zero_kernel(float* __restrict__ p, int n) {
    int idx = blockIdx.x * blockDim.x + threadIdx.x;
    if (idx < n) p[idx] = 0.0f;
}

__global__ void count_kernel(const int* __restrict__ dst, float* __restrict__ cnt) {
    int e = blockIdx.x * blockDim.x + threadIdx.x;
    if (e < NEDGES) atomicAdd(&cnt[dst[e]], 1.0f);
}

__global__ void invdenom_kernel(const float* __restrict__ cnt, float* __restrict__ inv) {
    int n = blockIdx.x * blockDim.x + threadIdx.x;
    if (n < NNODES) inv[n] = 1.0f / fmaxf(cnt[n], 1.0f);
}

// ---------------------------------------------------------------------------
// Fused GEMM: OUT[12288 x 96] = relu(H[12288 x 32]) @ [W1 | B1 | root]
// One wave per 16-row M tile; 6 N-tiles of 16; K=32 as 8 steps of K=4 fp32
// WMMA (V_WMMA_F32_16X16X4_F32). Weights (3 * 32x32 fp32 = 12 KB) staged in
// LDS once per block. All 256 lanes active => EXEC all-ones as WMMA requires.
//
// ISA layouts (§7.12.2):
//   A 16x4 f32 : lanes 0-15 -> {K=0 (v0), K=1 (v1)}, lanes 16-31 -> {K=2, K=3}
//   B 4x16 f32 : mirror of A with N in place of M
//   C/D 16x16  : VGPR j holds row j (lanes 0-15) / row j+8 (lanes 16-31)
// ---------------------------------------------------------------------------
__global__ __launch_bounds__(256)
void gemm_relu_wmma_kernel(const float* __restrict__ h,
                           const float* __restrict__ l1_w,   // W1 [32x32], [i*D+o]
                           const float* __restrict__ l1_b,   // B1 [32x32]
                           const float* __restrict__ root,   // root [32x32]
                           float* __restrict__ P1,
                           float* __restrict__ P2,
                           float* __restrict__ R) {
    __shared__ float wsh[3 * D * D];

    const int tid = threadIdx.x;
    for (int i = tid; i < 3 * D * D; i += 256) {
        float v;
        if (i < D * D)          v = l1_w[i];
        else if (i < 2 * D * D) v = l1_b[i - D * D];
        else                    v = root[i - 2 * D * D];
        wsh[i] = v;
    }
    __syncthreads();

    const int lane  = tid & 31;
    const int wave  = tid >> 5;
    const int m0    = (blockIdx.x * 8 + wave) * 16;   // 16-row tile base
    const int mrow  = m0 + (lane & 15);
    const int khalf = lane >> 4;                       // 0: K=k0,k0+1  1: K=k0+2,k0+3

    v8f acc[6] = {};

#pragma unroll
    for (int kk = 0; kk < 8; ++kk) {
        const int k0 = kk * 4 + khalf * 2;
        // A fragment: relu applied on load (reference relu's h before both
        // the message matmul and the root transform).
        v2f a;
        a.x = fmaxf(h[mrow * D + k0],     0.0f);
        a.y = fmaxf(h[mrow * D + k0 + 1], 0.0f);
#pragma unroll
        for (int t = 0; t < 6; ++t) {                  // 6 N-tiles: W1,W1,B1,B1,root,root
            const int base = (t >> 1) * (D * D);
            const int col  = (t & 1) * 16 + (lane & 15);
            v2f b;
            b.x = wsh[base + k0 * D + col];
            b.y = wsh[base + (k0 + 1) * D + col];
            acc[t] = __builtin_amdgcn_wmma_f32_16x16x4_f32(
                false, a, false, b, (short)0, acc[t], false, false);
        }
    }

    // Scatter C/D fragments: VGPR j -> row m0 + khalf*8 + j, col = lane&15.
    const int orow0 = m0 + khalf * 8;
    const int ocol  = lane & 15;
#pragma unroll
    for (int t = 0; t < 6; ++t) {
        float* out = (t < 2) ? P1 : ((t < 4) ? P2 : R);
        const int col = (t & 1) * 16 + ocol;
#pragma unroll
        for (int j = 0; j < 8; ++j)
            out[(orow0 + j) * D + col] = acc[t][j];
    }
}

// One wave per edge: lane d handles feature d.
// agg[dst] += ea[e]*P1[src] + P2[src]   (global_atomic_add_f32, coalesced rows)
__global__ __launch_bounds__(256)
void edge_scatter_kernel(const float* __restrict__ P1,
                         const float* __restrict__ P2,
                         const float* __restrict__ ea,
                         const int* __restrict__ src,
                         const int* __restrict__ dst,
                         float* __restrict__ agg) {
    const int lane   = threadIdx.x & 31;
    const int wave   = (blockIdx.x * blockDim.x + threadIdx.x) >> 5;
    const int nwaves = (gridDim.x * blockDim.x) >> 5;
    for (int e = wave; e < NEDGES; e += nwaves) {
        const int   s = src[e];
        const int   d = dst[e];
        const float w = ea[e];
        const float v = fmaf(w, P1[s * D + lane], P2[s * D + lane]);
        atomicAdd(&agg[d * D + lane], v);
    }
}

// h_next = agg * inv_denom + R + conv_bias
__global__ void finalize_kernel(const float* __restrict__ agg,
                                const float* __restrict__ inv,
                                const float* __restrict__ R,
                                const float* __restrict__ cbias,
                                float* __restrict__ hout) {
    int idx = blockIdx.x * blockDim.x + threadIdx.x;
    if (idx < NNODES * D) {
        int n = idx >> 5, d = idx & 31;
        hout[idx] = fmaf(agg[idx], inv[n], R[idx] + cbias[d]);
    }
}

extern "C" void kernel_launch(void* const* d_in, const int* in_sizes, int n_in,
                              void* d_out, int out_size, void* d_ws, size_t ws_size,
                              hipStream_t stream) {
    (void)in_sizes; (void)n_in; (void)out_size; (void)ws_size;

    const float* x     = (const float*)d_in[0];
    const float* ea    = (const float*)d_in[1];
    const float* l1_w  = (const float*)d_in[2];
    const float* l1_b  = (const float*)d_in[3];
    const float* ne_w  = (const float*)d_in[4];
    const float* ne_b  = (const float*)d_in[5];
    const float* root  = (const float*)d_in[6];
    const float* cbias = (const float*)d_in[7];
    const int*   ei    = (const int*)d_in[8];   // [2, E]
    const int*   src   = ei;
    const int*   dst   = ei + NEDGES;

    // Workspace layout (floats): 2 h ping-pong + P1,P2,R + agg + cnt + inv
    float* ws  = (float*)d_ws;
    float* h0  = ws;
    float* h1  = h0  + NNODES * D;
    float* P1  = h1  + NNODES * D;
    float* P2  = P1  + NNODES * D;
    float* R   = P2  + NNODES * D;
    float* agg = R   + NNODES * D;
    float* cnt = agg + NNODES * D;
    float* inv = cnt + NNODES;

    const int TPB = 256;
    const int ndBlocks = (NNODES * D + TPB - 1) / TPB;   // 1536
    const int nBlocks  = (NNODES + TPB - 1) / TPB;       // 48
    const int eBlocks  = (NEDGES + TPB - 1) / TPB;       // 512

    // Node embedding + (per-call) degree counts and reciprocal denominators.
    embed_kernel<<<ndBlocks, TPB, 0, stream>>>(x, ne_w, ne_b, h0);
    zero_kernel<<<nBlocks, TPB, 0, stream>>>(cnt, NNODES);
    count_kernel<<<eBlocks, TPB, 0, stream>>>(dst, cnt);
    invdenom_kernel<<<nBlocks, TPB, 0, stream>>>(cnt, inv);

    float* hcur  = h0;
    float* hnext = h1;
    for (int l = 0; l < NLAYERS; ++l) {
        // Fused relu + [P1|P2|R] GEMM: 96 blocks * 8 waves = 768 M-tiles.
        gemm_relu_wmma_kernel<<<NNODES / 128, TPB, 0, stream>>>(
            hcur, l1_w, l1_b, root, P1, P2, R);
        zero_kernel<<<ndBlocks, TPB, 0, stream>>>(agg, NNODES * D);
        edge_scatter_kernel<<<512, TPB, 0, stream>>>(P1, P2, ea, src, dst, agg);
        float* out = (l == NLAYERS - 1) ? (float*)d_out : hnext;
        finalize_kernel<<<ndBlocks, TPB, 0, stream>>>(agg, inv, R, cbias, out);
        float* t = hcur; hcur = hnext; hnext = t;
    }
}